// UniGINConv_34368328303048
// MI455X (gfx1250) — compile-verified
//
#include <hip/hip_runtime.h>

typedef float v2f __attribute__((ext_vector_type(2)));
typedef float v8f __attribute__((ext_vector_type(8)));

#define N_NODES 50000
#define N_EDGES 20000
#define N_INC   800000
#define D       128

// ---- workspace layout (floats) ----
// esum/Xe : [0, N_EDGES*D)                      = 2,560,000
// cnt     : [2,560,000, 2,580,000)              =    20,000
// Xv      : [2,580,000, 2,580,000 + N_NODES*D)  = 6,400,000
#define WS_ESUM 0
#define WS_CNT  (N_EDGES * D)
#define WS_XV   (N_EDGES * D + N_EDGES)
#define WS_TOTAL (N_EDGES * D + N_EDGES + N_NODES * D)   // 8,980,000 floats (mult of 4)

__global__ __launch_bounds__(256) void zero_ws_kernel(float4* __restrict__ ws, int n4) {
    int i = blockIdx.x * blockDim.x + threadIdx.x;
    if (i < n4) ws[i] = make_float4(0.f, 0.f, 0.f, 0.f);
}

// one wave (32 lanes) per incidence; lane owns 4 consecutive dims
__global__ __launch_bounds__(256) void scatter_edges_kernel(
    const float* __restrict__ X, const int* __restrict__ vertex,
    const int* __restrict__ edges, float* __restrict__ esum,
    float* __restrict__ cnt) {
    int gid  = blockIdx.x * blockDim.x + threadIdx.x;
    int inc  = gid >> 5;
    int lane = gid & 31;
    if (inc >= N_INC) return;
    int v = vertex[inc];
    int e = edges[inc];
    const float4 x = *(const float4*)(X + (size_t)v * D + lane * 4);
    float* dst = esum + (size_t)e * D + lane * 4;
    __hip_atomic_fetch_add(dst + 0, x.x, __ATOMIC_RELAXED, __HIP_MEMORY_SCOPE_AGENT);
    __hip_atomic_fetch_add(dst + 1, x.y, __ATOMIC_RELAXED, __HIP_MEMORY_SCOPE_AGENT);
    __hip_atomic_fetch_add(dst + 2, x.z, __ATOMIC_RELAXED, __HIP_MEMORY_SCOPE_AGENT);
    __hip_atomic_fetch_add(dst + 3, x.w, __ATOMIC_RELAXED, __HIP_MEMORY_SCOPE_AGENT);
    if (lane == 0)
        __hip_atomic_fetch_add(cnt + e, 1.0f, __ATOMIC_RELAXED, __HIP_MEMORY_SCOPE_AGENT);
}

// esum -> segment mean (in place)
__global__ __launch_bounds__(256) void edge_mean_kernel(
    float4* __restrict__ esum4, const float* __restrict__ cnt) {
    int i = blockIdx.x * blockDim.x + threadIdx.x;
    const int n4 = N_EDGES * (D / 4);
    if (i >= n4) return;
    int row = i / (D / 4);
    float s = 1.0f / fmaxf(cnt[row], 1.0f);
    float4 v = esum4[i];
    v.x *= s; v.y *= s; v.z *= s; v.w *= s;
    esum4[i] = v;
}

// one wave per incidence: Xv[vertex] += Xe[edges]
__global__ __launch_bounds__(256) void scatter_nodes_kernel(
    const float* __restrict__ Xe, const int* __restrict__ vertex,
    const int* __restrict__ edges, float* __restrict__ Xv) {
    int gid  = blockIdx.x * blockDim.x + threadIdx.x;
    int inc  = gid >> 5;
    int lane = gid & 31;
    if (inc >= N_INC) return;
    int v = vertex[inc];
    int e = edges[inc];
    const float4 x = *(const float4*)(Xe + (size_t)e * D + lane * 4);
    float* dst = Xv + (size_t)v * D + lane * 4;
    __hip_atomic_fetch_add(dst + 0, x.x, __ATOMIC_RELAXED, __HIP_MEMORY_SCOPE_AGENT);
    __hip_atomic_fetch_add(dst + 1, x.y, __ATOMIC_RELAXED, __HIP_MEMORY_SCOPE_AGENT);
    __hip_atomic_fetch_add(dst + 2, x.z, __ATOMIC_RELAXED, __HIP_MEMORY_SCOPE_AGENT);
    __hip_atomic_fetch_add(dst + 3, x.w, __ATOMIC_RELAXED, __HIP_MEMORY_SCOPE_AGENT);
}

// branch-free nan/inf policy: nan->0, +inf->100, -inf->-100
// (lowers to v_cmp + v_cndmask, no exec-mask churn)
__device__ __forceinline__ float clamp_nan_inf(float v) {
    float r = v;
    r = (v ==  __builtin_inff()) ?  100.0f : r;
    r = (v == -__builtin_inff()) ? -100.0f : r;
    r = (v != v)                 ?    0.0f : r;
    return r;
}

// fused: Xc = (1+eps)*X + Xv ; out = Xc @ W^T via V_WMMA_F32_16X16X4_F32
// one wave per 16x16 output tile; grid = (3125, 8)
__global__ __launch_bounds__(32) void gemm_wmma_kernel(
    const float* __restrict__ X, const float* __restrict__ W,
    const float* __restrict__ epsp, const float* __restrict__ Xv,
    float* __restrict__ out) {
    const int m0   = blockIdx.x * 16;
    const int n0   = blockIdx.y * 16;
    const int lane = threadIdx.x;
    const int r    = lane & 15;
    const int h    = lane >> 4;          // half-wave selects K pair / M half
    const float ope = 1.0f + epsp[0];

    const float* xrow = X  + (size_t)(m0 + r) * D;
    const float* vrow = Xv + (size_t)(m0 + r) * D;
    const float* wrow = W  + (size_t)(n0 + r) * D;   // B[k][n] = W[n][k]

    v8f acc = {};
    #pragma unroll 4
    for (int kt = 0; kt < D / 4; ++kt) {
        const int k = kt * 4 + 2 * h;    // this half-wave's K pair {k, k+1}
        // A fragment: Xc[m0+r][k..k+1]
        const float2 xa = *(const float2*)(xrow + k);
        const float2 xb = *(const float2*)(vrow + k);
        v2f a; a.x = ope * xa.x + xb.x;
               a.y = ope * xa.y + xb.y;
        // B fragment: W[n0+r][k..k+1]
        const v2f b = *(const v2f*)(wrow + k);
        acc = __builtin_amdgcn_wmma_f32_16x16x4_f32(
            /*neg_a=*/false, a, /*neg_b=*/false, b,
            /*c_mod=*/(short)0, acc, /*reuse_a=*/false, /*reuse_b=*/false);
    }

    // D layout: acc[j] = out[m0 + j + 8*h][n0 + r]
    float* orow = out + (size_t)(m0 + 8 * h) * D + n0 + r;
    #pragma unroll
    for (int j = 0; j < 8; ++j) {
        orow[(size_t)j * D] = clamp_nan_inf(acc[j]);
    }
}

extern "C" void kernel_launch(void* const* d_in, const int* in_sizes, int n_in,
                              void* d_out, int out_size, void* d_ws, size_t ws_size,
                              hipStream_t stream) {
    const float* X      = (const float*)d_in[0];   // 50000*128
    const float* W      = (const float*)d_in[1];   // 128*128
    const float* eps    = (const float*)d_in[2];   // 1
    const int*   vertex = (const int*)d_in[3];     // 800000
    const int*   edges  = (const int*)d_in[4];     // 800000

    float* ws   = (float*)d_ws;
    float* esum = ws + WS_ESUM;   // becomes Xe in place
    float* cnt  = ws + WS_CNT;
    float* Xv   = ws + WS_XV;
    float* out  = (float*)d_out;

    // 1) zero accumulators (esum, cnt, Xv)
    {
        const int n4 = WS_TOTAL / 4;
        zero_ws_kernel<<<(n4 + 255) / 256, 256, 0, stream>>>((float4*)ws, n4);
    }
    // 2) scatter X rows into edge sums + counts (wave per incidence)
    {
        const long long threads = (long long)N_INC * 32;
        scatter_edges_kernel<<<(int)((threads + 255) / 256), 256, 0, stream>>>(
            X, vertex, edges, esum, cnt);
    }
    // 3) segment mean in place
    {
        const int n4 = N_EDGES * (D / 4);
        edge_mean_kernel<<<(n4 + 255) / 256, 256, 0, stream>>>((float4*)esum, cnt);
    }
    // 4) scatter edge means into node sums
    {
        const long long threads = (long long)N_INC * 32;
        scatter_nodes_kernel<<<(int)((threads + 255) / 256), 256, 0, stream>>>(
            esum, vertex, edges, Xv);
    }
    // 5) fused (1+eps)X + Xv, GEMM with W^T via f32 WMMA, nan/inf clamp
    {
        dim3 grid(N_NODES / 16, D / 16);   // 3125 x 8
        gemm_wmma_kernel<<<grid, 32, 0, stream>>>(X, W, eps, Xv, out);
    }
}